// CapsuleLayer_62706522521966
// MI455X (gfx1250) — compile-verified
//
#include <hip/hip_runtime.h>
#include <hip/hip_bf16.h>

// CapsuleLayer on MI455X (gfx1250).
// Routing coefficients c = softmax over a singleton axis == 1 always, so the
// whole layer collapses to:
//   s[b, o*32+u] = sum_{i,c} x[b,i,c] * w[i,o,c,u]   (GEMM: M=32, N=1024, K=32768)
//   out = squash(s) over u (per (b,o) row of 32)
// Memory-bound: 132 MB reads @ 23.3 TB/s ~= 5.7us, needing ~375 TF fp32 matrix
// throughput -> use V_WMMA_F32_16X16X4_F32 (full fp32 fidelity).

typedef __attribute__((ext_vector_type(2))) float v2f;
typedef __attribute__((ext_vector_type(8))) float v8f;

#define B_SZ   32
#define I_CAP  2048
#define C_DIM  16
#define O_CAP  32
#define U_DIM  32
#define K_TOT  (I_CAP * C_DIM)      // 32768
#define N_TOT  (O_CAP * U_DIM)      // 1024
#define W_ISTRIDE (O_CAP * C_DIM * U_DIM)  // 16384 floats per i in w
#define NSPLIT 32                   // K-splits per N-tile
#define I_PER_WAVE (I_CAP / NSPLIT) // 64 i-values per wave

// One pipeline stage: data for 4 WMMA k-steps (K=16, i.e. one full 'i').
struct Stage {
    v2f   a0[4];   // A tile (m-tile 0), per k-step: lane holds A[m, k0+2*half .. +1]
    v2f   a1[4];   // A tile (m-tile 1)
    float b0[4];   // B[k0 + 2*half + 0, n]
    float b1[4];   // B[k0 + 2*half + 1, n]
};

__device__ __forceinline__ void load_stage(Stage& s,
                                           const float* __restrict__ a0,
                                           const float* __restrict__ a1,
                                           const float* __restrict__ b) {
#pragma unroll
    for (int ks = 0; ks < 4; ++ks) {
        const int c0 = ks * 4;                     // k-step base within C_DIM
        s.a0[ks] = *(const v2f*)(a0 + c0);         // global_load_b64
        s.a1[ks] = *(const v2f*)(a1 + c0);         // global_load_b64
        s.b0[ks] = b[c0 * U_DIM];                  // global_load_b32 (coalesced over u)
        s.b1[ks] = b[c0 * U_DIM + U_DIM];
    }
}

__device__ __forceinline__ void consume(const Stage& s, v8f& acc0, v8f& acc1) {
#pragma unroll
    for (int ks = 0; ks < 4; ++ks) {
        v2f b = { s.b0[ks], s.b1[ks] };
        acc0 = __builtin_amdgcn_wmma_f32_16x16x4_f32(false, s.a0[ks], false, b,
                                                     (short)0, acc0, false, false);
        acc1 = __builtin_amdgcn_wmma_f32_16x16x4_f32(false, s.a1[ks], false, b,
                                                     (short)0, acc1, false, false);
    }
}

__global__ void caps_zero(float* __restrict__ out) {
    int idx = blockIdx.x * blockDim.x + threadIdx.x;
    if (idx < B_SZ * N_TOT) out[idx] = 0.0f;
}

// grid = (64 N-tiles, 4), block = 256 (8 waves). Wave = one (N-tile, K-split).
__global__ void caps_gemm(const float* __restrict__ X,   // [32][32768]  (b, i*16+c)
                          const float* __restrict__ W,   // [2048][32][16][32] (i,o,c,u)
                          float* __restrict__ S) {       // [32][1024] accumulators
    const int lane  = threadIdx.x & 31;
    const int wave  = threadIdx.x >> 5;
    const int ntile = blockIdx.x;                    // 0..63
    const int ksplt = blockIdx.y * 8 + wave;         // 0..31
    const int kh    = lane >> 4;                     // half-wave (K sub-offset)
    const int lm    = lane & 15;
    const int o     = ntile >> 1;
    const int u     = ((ntile & 1) << 4) + lm;       // u index within o
    const int n     = (ntile << 4) + lm;             // global column 0..1023
    const int i0    = ksplt * I_PER_WAVE;

    // A lane base: X[b = lm (+16 per m-tile), k = i*16 + c0 + 2*kh + {0,1}]
    const float* aBase0 = X + (size_t)lm * K_TOT + (size_t)i0 * C_DIM + 2 * kh;
    const float* aBase1 = aBase0 + (size_t)16 * K_TOT;
    // B lane base: W[i*16384 + o*512 + (c0 + 2*kh + j)*32 + u]
    const float* bBase  = W + (size_t)i0 * W_ISTRIDE + o * (C_DIM * U_DIM)
                            + kh * (2 * U_DIM) + u;

    v8f acc0 = {0.f,0.f,0.f,0.f,0.f,0.f,0.f,0.f};
    v8f acc1 = {0.f,0.f,0.f,0.f,0.f,0.f,0.f,0.f};

    Stage s0, s1;
    load_stage(s0, aBase0, aBase1, bBase);

#pragma unroll 1
    for (int ii = 0; ii < I_PER_WAVE; ii += 2) {
        // stream prefetch on the 128MB w tensor (global_prefetch_b8)
        if (ii + 8 < I_PER_WAVE)
            __builtin_prefetch(bBase + (size_t)(ii + 8) * W_ISTRIDE, 0, 0);

        load_stage(s1, aBase0 + (ii + 1) * C_DIM,
                       aBase1 + (ii + 1) * C_DIM,
                       bBase + (size_t)(ii + 1) * W_ISTRIDE);
        consume(s0, acc0, acc1);                     // 8x v_wmma_f32_16x16x4_f32

        if (ii + 2 < I_PER_WAVE)
            load_stage(s0, aBase0 + (ii + 2) * C_DIM,
                           aBase1 + (ii + 2) * C_DIM,
                           bBase + (size_t)(ii + 2) * W_ISTRIDE);
        consume(s1, acc0, acc1);
    }

    // C/D layout: VGPR r -> M = r (lanes 0-15) / r+8 (lanes 16-31); N = lane%16.
    const int mbase = kh * 8;
#pragma unroll
    for (int r = 0; r < 8; ++r) {
        unsafeAtomicAdd(S + (size_t)(mbase + r) * N_TOT + n, acc0[r]);        // global_atomic_add_f32
        unsafeAtomicAdd(S + (size_t)(16 + mbase + r) * N_TOT + n, acc1[r]);
    }
}

// In-place squash: one wave32 per (b, o) row of 32 u-values.
__global__ void caps_squash(float* __restrict__ S) {
    const int gw   = (blockIdx.x * blockDim.x + threadIdx.x) >> 5;
    const int lane = threadIdx.x & 31;
    if (gw >= B_SZ * O_CAP) return;
    const int b = gw >> 5;
    const int o = gw & 31;
    float* p = S + (size_t)b * N_TOT + o * U_DIM;
    float v  = p[lane];
    float sq = v * v;
#pragma unroll
    for (int off = 16; off > 0; off >>= 1)
        sq += __shfl_xor(sq, off, 32);               // wave32 butterfly
    float nrm = sqrtf(sq);
    p[lane] = v * (nrm / (1.0f + sq));               // v * |v| / (1 + |v|^2)
}

extern "C" void kernel_launch(void* const* d_in, const int* in_sizes, int n_in,
                              void* d_out, int out_size, void* d_ws, size_t ws_size,
                              hipStream_t stream) {
    const float* X = (const float*)d_in[0];   // [32, 2048, 16]
    const float* W = (const float*)d_in[1];   // [2048, 32, 16, 32]
    float* out = (float*)d_out;               // [32, 32, 1, 32] == s matrix layout

    caps_zero<<<dim3((B_SZ * N_TOT + 255) / 256), dim3(256), 0, stream>>>(out);
    caps_gemm<<<dim3(64, 4), dim3(256), 0, stream>>>(X, W, out);
    caps_squash<<<dim3((B_SZ * O_CAP * 32 + 255) / 256), dim3(256), 0, stream>>>(out);
}